// AttentionLinearBiases_27350351741345
// MI455X (gfx1250) — compile-verified
//
#include <hip/hip_runtime.h>
#include <hip/hip_bf16.h>

// ALiBi causal bias add: out[b,h,i,j] = in[b,h,i,j] + slope(h) * (j - (SEQ-1))
// Shapes fixed by the reference: (2, 16, 2048, 2048) float32.
//
// Memory-bound (0.125 FLOP/byte): optimize purely for streaming bandwidth on
// MI455X — b128 vector memory ops, non-temporal cache policy (no reuse,
// 1 GiB >> 192 MB L2), wave32-sized blocks, 64 B per thread for MLP.

typedef float v4f __attribute__((ext_vector_type(4)));

#define SEQ        2048
#define ROWQ       (SEQ / 4)       // float4 per row = 512
#define HEADS      16
#define PERMAT_LOG 20              // (2048*2048/4) = 1<<20 float4 per head-matrix
#define TPB        256             // 8 waves (wave32)
#define V4_PER_THR 4               // 4 x float4 = 64 B per thread

__global__ __launch_bounds__(TPB) void alibi_add_kernel(
    const v4f* __restrict__ in, v4f* __restrict__ out, unsigned int total4)
{
    // Block tile: TPB * V4_PER_THR contiguous float4, thread-interleaved for
    // perfectly coalesced b128 accesses.
    unsigned int base = blockIdx.x * (TPB * V4_PER_THR) + threadIdx.x;

#pragma unroll
    for (int k = 0; k < V4_PER_THR; ++k) {
        unsigned int idx = base + (unsigned int)k * TPB;
        if (idx < total4) {
            // Recover column (in float4 units) and head from the flat index.
            unsigned int j4 = idx & (ROWQ - 1u);
            unsigned int h  = (idx >> PERMAT_LOG) & (HEADS - 1u);

            // slope(h) = 2^(-(h+1)/2), exact via ldexp:
            //   e = h+1 even : 2^(-e/2)
            //   e = h+1 odd  : 2^(-1/2) * 2^(-(e-1)/2)
            unsigned int e = h + 1u;
            float mant  = (e & 1u) ? 0.70710678118654752440f : 1.0f;
            float slope = ldexpf(mant, -(int)(e >> 1));

            // distance for first lane-column of this float4: j - (SEQ-1)
            float col = (float)(j4 * 4u) - (float)(SEQ - 1);

            // Streaming: non-temporal b128 load/store (TH=NT), FMA the bias.
            v4f a = __builtin_nontemporal_load(&in[idx]);
            v4f r;
            r.x = __builtin_fmaf(slope, col,        a.x);
            r.y = __builtin_fmaf(slope, col + 1.0f, a.y);
            r.z = __builtin_fmaf(slope, col + 2.0f, a.z);
            r.w = __builtin_fmaf(slope, col + 3.0f, a.w);
            __builtin_nontemporal_store(r, &out[idx]);
        }
    }
}

extern "C" void kernel_launch(void* const* d_in, const int* in_sizes, int n_in,
                              void* d_out, int out_size, void* d_ws, size_t ws_size,
                              hipStream_t stream) {
    const v4f* in  = (const v4f*)d_in[0];
    v4f*       out = (v4f*)d_out;

    unsigned int total4 = (unsigned int)(in_sizes[0] / 4);     // 33,554,432
    unsigned int per_block = TPB * V4_PER_THR;                 // 1024 float4
    unsigned int blocks = (total4 + per_block - 1u) / per_block; // 32,768

    alibi_add_kernel<<<blocks, TPB, 0, stream>>>(in, out, total4);
}